// PriorWeightPhaseType_22789096472967
// MI455X (gfx1250) — compile-verified
//
#include <hip/hip_runtime.h>
#include <hip/hip_bf16.h>
#include <math.h>

// Phase-type log-density sum on MI455X (gfx1250, wave32).
//
// density(w) = alpha^T expm(w*S) s,  s = -S @ 1 ;  out = sum_m log(density)
//
// expmv: u <- [exp(w S^T / P)]^P alpha, each stage an order-Q Taylor series of
// S^T matvecs, batched 32 weights per wave on v_wmma_f32_16x16x4_f32.
//
// Layout trick (zero-relayout pipeline):
//   Each lane owns the full 8-component state of ONE weight (lane = weight).
//   The state registers t[0..7] are fed directly as the B operand:
//   b(s,i) = t[2s+i]. Hardware places low lanes at B rows {0,1,4,5,8,9,12,13}
//   (group A) and high lanes at rows {2,3,6,7,10,11,14,15} (group B).
//   The constant A operand A' scatters S^T into exactly those column slots:
//     rows m<8  (group-A outputs) read only low-k slots,
//     rows m>=8 (group-B outputs) read only high-k slots.
//   So A' @ B_eff == blockdiag(S^T,S^T) applied to all 32 per-lane states,
//   and the WMMA D output lands back in the per-lane layout directly:
//   no shuffles, no selects, no LDS between Taylor steps.

typedef __attribute__((ext_vector_type(2))) float v2f;
typedef __attribute__((ext_vector_type(8))) float v8f;

#define PT_STAGES 8   // exp(A) = exp(A/8)^8
#define PT_ORDER  9   // Taylor order per stage (per-stage norm <= ~0.8)

__global__ __launch_bounds__(256) void pt_logdens_kernel(
    const float* __restrict__ w, const float* __restrict__ S,
    const float* __restrict__ alpha, float* __restrict__ partial, int M) {
  const int tid  = blockIdx.x * 256 + threadIdx.x;
  const int lane = threadIdx.x & 31;
  const int m    = lane & 15;
  const bool low = (lane & 16) == 0;

  // Per-lane weight; out-of-range lanes compute a benign w=1 (masked later).
  const float wl = (tid < M) ? w[tid] : 1.0f;
  const float wp = wl * (1.0f / (float)PT_STAGES);

  // Constant A operand A' (16x16): lane holds element (m, k) with
  // k = 4s + kh + i, kh = low?0:2, for slices s=0..3, i=0..1.
  // Slot (s,i) carries S^T column j = 2s+i:
  //   low-k slots  -> rows m<8  : A'[m,k] = S^T[m, j]   = S[j*8 + m]
  //   high-k slots -> rows m>=8 : A'[m,k] = S^T[m-8, j] = S[j*8 + m-8]
  const int mm = m & 7;
  const bool active = (low && m < 8) || (!low && m >= 8);
  v2f av[4];
#pragma unroll
  for (int s = 0; s < 4; ++s) {
    float e[2];
#pragma unroll
    for (int i = 0; i < 2; ++i) {
      const int j = 2 * s + i;
      e[i] = active ? S[j * 8 + mm] : 0.0f;
    }
    av[s].x = e[0];
    av[s].y = e[1];
  }

  // Exit-rate vector: sv[i] = -sum_j S[i][j]  (uniform -> scalar loads)
  float sv[8];
#pragma unroll
  for (int r = 0; r < 8; ++r) {
    float acc = 0.0f;
#pragma unroll
    for (int j = 0; j < 8; ++j) acc += S[r * 8 + j];
    sv[r] = -acc;
  }

  // State u: component r of this lane's weight's vector; t = current term.
  float u[8], t[8];
#pragma unroll
  for (int r = 0; r < 8; ++r) u[r] = alpha[r];

#pragma unroll 1
  for (int stage = 0; stage < PT_STAGES; ++stage) {
#pragma unroll
    for (int r = 0; r < 8; ++r) t[r] = u[r];  // Taylor restarts from u
#pragma unroll
    for (int k = 1; k <= PT_ORDER; ++k) {
      // acc = blockdiag(S^T,S^T) @ T via 4 chained K=4 fp32 WMMAs;
      // B operands are the raw per-lane term registers (zero relayout).
      v8f acc = {0.f, 0.f, 0.f, 0.f, 0.f, 0.f, 0.f, 0.f};
      v2f b0 = {t[0], t[1]};
      v2f b1 = {t[2], t[3]};
      v2f b2 = {t[4], t[5]};
      v2f b3 = {t[6], t[7]};
      acc = __builtin_amdgcn_wmma_f32_16x16x4_f32(false, av[0], false, b0,
                                                  (short)0, acc, false, false);
      acc = __builtin_amdgcn_wmma_f32_16x16x4_f32(false, av[1], false, b1,
                                                  (short)0, acc, false, false);
      acc = __builtin_amdgcn_wmma_f32_16x16x4_f32(false, av[2], false, b2,
                                                  (short)0, acc, false, false);
      acc = __builtin_amdgcn_wmma_f32_16x16x4_f32(false, av[3], false, b3,
                                                  (short)0, acc, false, false);
      // term_k = (w/(P*k)) * (S^T term_{k-1});  u += term_k
      const float c = wp * (1.0f / (float)k);
#pragma unroll
      for (int r = 0; r < 8; ++r) {
        t[r] = acc[r] * c;
        u[r] += t[r];
      }
    }
  }

  // density = u . s ; masked log
  float dens = 0.0f;
#pragma unroll
  for (int r = 0; r < 8; ++r) dens += u[r] * sv[r];
  float contrib = (tid < M) ? logf(dens) : 0.0f;

  // Block reduction (deterministic tree), one partial per block.
  __shared__ float red[256];
  red[threadIdx.x] = contrib;
  __syncthreads();
#pragma unroll
  for (int s = 128; s > 0; s >>= 1) {
    if (threadIdx.x < s) red[threadIdx.x] += red[threadIdx.x + s];
    __syncthreads();
  }
  if (threadIdx.x == 0) partial[blockIdx.x] = red[0];
}

__global__ __launch_bounds__(256) void pt_reduce_kernel(
    const float* __restrict__ partial, int n, float* __restrict__ out) {
  __shared__ float red[256];
  float acc = 0.0f;
  for (int i = threadIdx.x; i < n; i += 256) acc += partial[i];
  red[threadIdx.x] = acc;
  __syncthreads();
#pragma unroll
  for (int s = 128; s > 0; s >>= 1) {
    if (threadIdx.x < s) red[threadIdx.x] += red[threadIdx.x + s];
    __syncthreads();
  }
  if (threadIdx.x == 0) out[0] = red[0];
}

extern "C" void kernel_launch(void* const* d_in, const int* in_sizes, int n_in,
                              void* d_out, int out_size, void* d_ws, size_t ws_size,
                              hipStream_t stream) {
  const float* w     = (const float*)d_in[0];
  const float* S     = (const float*)d_in[1];
  const float* alpha = (const float*)d_in[2];
  float* out = (float*)d_out;
  float* partial = (float*)d_ws;

  const int M = in_sizes[0];
  const int nblk = (M + 255) / 256;

  pt_logdens_kernel<<<nblk, 256, 0, stream>>>(w, S, alpha, partial, M);
  pt_reduce_kernel<<<1, 256, 0, stream>>>(partial, nblk, out);
}